// GCNNet_67147518705920
// MI455X (gfx1250) — compile-verified
//
#include <hip/hip_runtime.h>

typedef __attribute__((ext_vector_type(2))) float v2f;
typedef __attribute__((ext_vector_type(8))) float v8f;

static constexpr int   Hh  = 64;      // hidden width
static constexpr float EPS = 1e-5f;

// ---------------- small utility kernels ----------------

__global__ void k_fill(float* p, int n, float v) {
    int i = blockIdx.x * blockDim.x + threadIdx.x;
    if (i < n) p[i] = v;
}

__global__ void k_deg_accum(const int* __restrict__ ei, int E, float* deg) {
    int e = blockIdx.x * blockDim.x + threadIdx.x;
    if (e >= E) return;
    int r = ei[e], c = ei[E + e];
    if (r != c) atomicAdd(&deg[r], 1.0f);
}

__global__ void k_rsqrt_inplace(float* p, int n) {
    int i = blockIdx.x * blockDim.x + threadIdx.x;
    if (i < n) p[i] = rsqrtf(p[i]);
}

// column-wise sum / sumsq; blockDim.x == ncols, each block covers rowsPerBlock rows
__global__ void k_bn_stats(const float* __restrict__ x, int nrows, int ncols,
                           int rowsPerBlock, float* __restrict__ stats) {
    int j  = threadIdx.x;
    int r0 = blockIdx.x * rowsPerBlock;
    int r1 = min(r0 + rowsPerBlock, nrows);
    float s = 0.f, ss = 0.f;
    for (int r = r0; r < r1; ++r) {
        float v = x[(long)r * ncols + j];
        s += v; ss += v * v;
    }
    atomicAdd(&stats[j], s);
    atomicAdd(&stats[ncols + j], ss);
}

// per-column BN affine:   xn = x*scale + shift
__global__ void k_bn_fold(const float* __restrict__ stats,
                          const float* __restrict__ g, const float* __restrict__ b,
                          int ncols, int nrows,
                          float* __restrict__ scale, float* __restrict__ shift) {
    int j = blockIdx.x * blockDim.x + threadIdx.x;
    if (j >= ncols) return;
    float inv_n = 1.0f / (float)nrows;
    float m    = stats[j] * inv_n;
    float var  = stats[ncols + j] * inv_n - m * m;
    float rstd = rsqrtf(var + EPS);
    float sc   = g[j] * rstd;
    scale[j] = sc;
    shift[j] = b[j] - m * sc;
}

// fold BN affine into weights:  Wp[k,h] = scale[k]*W[k,h];  cb[h] = sum_k shift[k]*W[k,h]
__global__ void k_fold_w(const float* __restrict__ W, const float* __restrict__ scale,
                         const float* __restrict__ shift, int K,
                         float* __restrict__ Wp, float* __restrict__ cb) {
    int h = threadIdx.x;   // blockDim == Hh
    float acc = 0.f;
    for (int k = 0; k < K; ++k) {
        float w = W[k * Hh + h];
        Wp[k * Hh + h] = scale[k] * w;
        acc += shift[k] * w;
    }
    cb[h] = acc;
}

// ---------------- WMMA GEMM:  out[N,64] = A[N,K] @ Wp[K,64] + cb (opt relu) ----------------
// 256 threads = 8 waves; block tile = 32 rows x 64 cols; each wave does one 16x16 tile.
template <int K, bool RELU>
__global__ __launch_bounds__(256) void k_gemm(const float* __restrict__ A,
                                              const float* __restrict__ Wp,
                                              const float* __restrict__ cb,
                                              float* __restrict__ out, int nrows) {
    __shared__ float As[32 * K];
    __shared__ float Bs[K * Hh];

    const int tid     = threadIdx.x;
    const int rowBase = blockIdx.x * 32;

    for (int i = tid; i < K * Hh; i += 256) Bs[i] = Wp[i];
    for (int i = tid; i < 32 * K; i += 256) {
        int r = rowBase + i / K;
        As[i] = (r < nrows) ? A[(long)rowBase * K + i] : 0.f;
    }
    __syncthreads();

    const int wid     = tid >> 5;
    const int lane    = tid & 31;
    const int lane_lo = lane & 15;
    const int half    = lane >> 4;           // 0: K pair {0,1}, 1: K pair {2,3}
    const int rowTile = (wid >> 2) * 16;     // 0 or 16
    const int colTile = (wid & 3) * 16;      // 0,16,32,48

    v8f acc = {0.f, 0.f, 0.f, 0.f, 0.f, 0.f, 0.f, 0.f};
    const float* as = &As[(rowTile + lane_lo) * K];

    for (int k0 = 0; k0 < K; k0 += 4) {
        int ka = k0 + half * 2;
        v2f a, b;
        // A 16x4 f32 layout: lanes 0-15 hold M=lane (K=k0,k0+1); lanes 16-31 (K=k0+2,k0+3)
        a.x = as[ka];
        a.y = as[ka + 1];
        // B 4x16 f32 layout: lanes 0-15 hold N=lane (K=k0,k0+1); lanes 16-31 (K=k0+2,k0+3)
        b.x = Bs[ka * Hh + colTile + lane_lo];
        b.y = Bs[(ka + 1) * Hh + colTile + lane_lo];
        acc = __builtin_amdgcn_wmma_f32_16x16x4_f32(false, a, false, b,
                                                    (short)0, acc, false, false);
    }

    // C/D layout: lanes 0-15 -> M = v, lanes 16-31 -> M = v+8; N = lane_lo
    float bias = cb[colTile + lane_lo];
    int   c    = colTile + lane_lo;
#pragma unroll
    for (int v = 0; v < 8; ++v) {
        int r = rowBase + rowTile + v + 8 * half;
        if (r < nrows) {
            float val = acc[v] + bias;
            if (RELU) val = fmaxf(val, 0.f);
            out[(long)r * Hh + c] = val;
        }
    }
}

// ---------------- graph aggregation ----------------

// self-loop contribution: agg[n,:] = dinv[n]^2 * hw[n,:]
__global__ void k_agg_self(const float* __restrict__ hw, const float* __restrict__ dinv,
                           float* __restrict__ agg, int n) {
    long i = (long)blockIdx.x * blockDim.x + threadIdx.x;
    if (i >= (long)n * Hh) return;
    int node = (int)(i >> 6);
    float d = dinv[node];
    agg[i] = d * d * hw[i];
}

// per edge: agg[c,:] += dinv[r]*dinv[c] * hw[r,:]   (64 lanes per edge)
__global__ void k_edge_scatter(const int* __restrict__ ei, int E,
                               const float* __restrict__ dinv,
                               const float* __restrict__ hw, float* __restrict__ agg) {
    long t = (long)blockIdx.x * blockDim.x + threadIdx.x;
    int  e = (int)(t >> 6);
    int  j = (int)(t & 63);
    if (e >= E) return;
    int r = ei[e], c = ei[E + e];
    if (r == c) return;                 // removed self-loops have weight 0
    float w = dinv[r] * dinv[c];
    atomicAdd(&agg[(long)c * Hh + j], w * hw[(long)r * Hh + j]);
}

__global__ void k_finalize_relu(const float* __restrict__ agg, const float* __restrict__ bias,
                                float* __restrict__ h, int n) {
    long i = (long)blockIdx.x * blockDim.x + threadIdx.x;
    if (i >= (long)n * Hh) return;
    int j = (int)(i & 63);
    h[i] = fmaxf(agg[i] + bias[j], 0.f);
}

__global__ void k_pool(const float* __restrict__ h, const int* __restrict__ batch,
                       float* __restrict__ hg, int n) {
    long i = (long)blockIdx.x * blockDim.x + threadIdx.x;
    if (i >= (long)n * Hh) return;
    int node = (int)(i >> 6), j = (int)(i & 63);
    atomicAdd(&hg[(long)batch[node] * Hh + j], h[i]);
}

// ---------------- head: BN -> fc+relu -> BN -> classifier (tiny, one block of 64) ----------------
__global__ void k_head(const float* __restrict__ hg_in, int G, int C,
                       const float* bn_fc_g, const float* bn_fc_b,
                       const float* W_fc, const float* b_fc,
                       const float* bn_h_g, const float* bn_h_b,
                       const float* W_cls, const float* b_cls,
                       float* __restrict__ out) {
    __shared__ float t0[32 * Hh];
    __shared__ float t1[32 * Hh];
    int h = threadIdx.x;                 // one thread per hidden column
    float s = 0.f, ss = 0.f;
    for (int g = 0; g < G; ++g) { float v = hg_in[g * Hh + h]; t0[g * Hh + h] = v; s += v; ss += v * v; }
    float m = s / G, var = ss / G - m * m, rstd = rsqrtf(var + EPS);
    float sc = bn_fc_g[h] * rstd, sh = bn_fc_b[h] - m * sc;
    for (int g = 0; g < G; ++g) t0[g * Hh + h] = t0[g * Hh + h] * sc + sh;
    __syncthreads();

    s = 0.f; ss = 0.f;
    for (int g = 0; g < G; ++g) {
        float acc = b_fc[h];
        for (int k = 0; k < Hh; ++k) acc += t0[g * Hh + k] * W_fc[k * Hh + h];
        acc = fmaxf(acc, 0.f);
        t1[g * Hh + h] = acc;
        s += acc; ss += acc * acc;
    }
    __syncthreads();
    m = s / G; var = ss / G - m * m; rstd = rsqrtf(var + EPS);
    sc = bn_h_g[h] * rstd; sh = bn_h_b[h] - m * sc;
    for (int g = 0; g < G; ++g) t1[g * Hh + h] = t1[g * Hh + h] * sc + sh;
    __syncthreads();

    if (h < C) {
        for (int g = 0; g < G; ++g) {
            float acc = b_cls[h];
            for (int k = 0; k < Hh; ++k) acc += t1[g * Hh + k] * W_cls[k * C + h];
            out[g * C + h] = acc;
        }
    }
}

// ---------------- host launcher ----------------

extern "C" void kernel_launch(void* const* d_in, const int* in_sizes, int n_in,
                              void* d_out, int out_size, void* d_ws, size_t ws_size,
                              hipStream_t stream) {
    (void)n_in; (void)ws_size;
    const float* x         = (const float*)d_in[0];
    const int*   ei        = (const int*)  d_in[1];
    const int*   batch     = (const int*)  d_in[2];
    const float* W_feat    = (const float*)d_in[4];
    const float* W_conv    = (const float*)d_in[5];
    const float* b_conv    = (const float*)d_in[6];
    const float* W_fc      = (const float*)d_in[7];
    const float* b_fc      = (const float*)d_in[8];
    const float* W_cls     = (const float*)d_in[9];
    const float* b_cls     = (const float*)d_in[10];
    const float* bn_feat_g = (const float*)d_in[11];
    const float* bn_feat_b = (const float*)d_in[12];
    const float* bn_conv_g = (const float*)d_in[13];
    const float* bn_conv_b = (const float*)d_in[14];
    const float* bn_fc_g   = (const float*)d_in[15];
    const float* bn_fc_b   = (const float*)d_in[16];
    const float* bn_h_g    = (const float*)d_in[17];
    const float* bn_h_b    = (const float*)d_in[18];

    const int F = 128;
    const int N = in_sizes[0] / F;
    const int E = in_sizes[1] / 2;
    const int C = in_sizes[10];          // b_cls length
    const int G = out_size / C;

    // workspace carve-out
    char* p = (char*)d_ws;
    auto take = [&](size_t bytes) -> void* {
        void* q = (void*)p;
        p += (bytes + 255) & ~(size_t)255;
        return q;
    };
    float* dinv  = (float*)take((size_t)N * 4);
    float* stats = (float*)take(2 * F * 4);
    float* scale = (float*)take(F * 4);
    float* shift = (float*)take(F * 4);
    float* Wp    = (float*)take((size_t)F * Hh * 4);
    float* cb    = (float*)take(Hh * 4);
    float* hg    = (float*)take((size_t)G * Hh * 4);
    float* bufH  = (float*)take((size_t)N * Hh * 4);
    float* bufW  = (float*)take((size_t)N * Hh * 4);
    float* bufA  = (float*)take((size_t)N * Hh * 4);

    const int ROWS = 256;                       // rows per bn_stats block
    long NH = (long)N * Hh;

    // ---- gcn_norm: deg (self-loop = 1) -> dinv = rsqrt(deg) ----
    k_fill<<<(N + 255) / 256, 256, 0, stream>>>(dinv, N, 1.0f);
    k_deg_accum<<<(E + 255) / 256, 256, 0, stream>>>(ei, E, dinv);
    k_rsqrt_inplace<<<(N + 255) / 256, 256, 0, stream>>>(dinv, N);

    // ---- feat stage: BN(x) folded into W_feat, then GEMM + relu ----
    k_fill<<<1, 256, 0, stream>>>(stats, 2 * F, 0.0f);
    k_bn_stats<<<(N + ROWS - 1) / ROWS, F, 0, stream>>>(x, N, F, ROWS, stats);
    k_bn_fold<<<1, F, 0, stream>>>(stats, bn_feat_g, bn_feat_b, F, N, scale, shift);
    k_fold_w<<<1, Hh, 0, stream>>>(W_feat, scale, shift, F, Wp, cb);
    k_gemm<128, true><<<(N + 31) / 32, 256, 0, stream>>>(x, Wp, cb, bufH, N);

    // ---- 3 GCN conv layers ----
    for (int i = 0; i < 3; ++i) {
        k_fill<<<1, 128, 0, stream>>>(stats, 2 * Hh, 0.0f);
        k_bn_stats<<<(N + ROWS - 1) / ROWS, Hh, 0, stream>>>(bufH, N, Hh, ROWS, stats);
        k_bn_fold<<<1, Hh, 0, stream>>>(stats, bn_conv_g + i * Hh, bn_conv_b + i * Hh,
                                        Hh, N, scale, shift);
        k_fold_w<<<1, Hh, 0, stream>>>(W_conv + (size_t)i * Hh * Hh, scale, shift, Hh, Wp, cb);
        k_gemm<64, false><<<(N + 31) / 32, 256, 0, stream>>>(bufH, Wp, cb, bufW, N);

        k_agg_self<<<(int)((NH + 255) / 256), 256, 0, stream>>>(bufW, dinv, bufA, N);
        k_edge_scatter<<<(int)(((long)E * Hh + 255) / 256), 256, 0, stream>>>(ei, E, dinv, bufW, bufA);
        k_finalize_relu<<<(int)((NH + 255) / 256), 256, 0, stream>>>(bufA, b_conv + i * Hh, bufH, N);
    }

    // ---- global add pool + head ----
    k_fill<<<(G * Hh + 255) / 256, 256, 0, stream>>>(hg, G * Hh, 0.0f);
    k_pool<<<(int)((NH + 255) / 256), 256, 0, stream>>>(bufH, batch, hg, N);
    k_head<<<1, Hh, 0, stream>>>(hg, G, C, bn_fc_g, bn_fc_b, W_fc, b_fc,
                                 bn_h_g, bn_h_b, W_cls, b_cls, (float*)d_out);
}